// ComplexGroupedMLP_70592082477333
// MI455X (gfx1250) — compile-verified
//
#include <hip/hip_runtime.h>
#include <hip/hip_bf16.h>

// ComplexGroupedMLP: y[b,m,g,o] = sum_i x[b,m,g,i] * w[g,i,o]  (complex fp32),
// out stored as (..., o, 2) interleaved real/imag.
// Shapes: BATCH*MODES = 16384 rows, GROUPS=32, BLOCK=64.
//
// Gauss 3-mult complex GEMM on V_WMMA_F32_16X16X4_F32:
//   P1 = Xr*Wr, P2 = Xi*Wi, P3 = (Xr+Xi)*(Wr+Wi)
//   Yr = P1 - P2 ; Yi = P3 - P1 - P2
// Weight fragments pre-deinterleaved into LDS once per block (one group/block).

typedef __attribute__((ext_vector_type(2))) float v2f;
typedef __attribute__((ext_vector_type(8))) float v8f;

#define CG_ROWS    16384
#define CG_GROUPS  32
#define CG_BLK     64
#define CG_RSTRIDE (CG_GROUPS * CG_BLK)   // 2048 floats per flattened (b,m) row

__global__ __launch_bounds__(256)
void ComplexGroupedMLP_70592082477333_kernel(const float* __restrict__ xr,
                                             const float* __restrict__ xi,
                                             const float* __restrict__ w,
                                             float* __restrict__ out)
{
    // Fragment-ready weight tiles: [ks][nt][lane] -> v2f (K-pair for this lane).
    __shared__ v2f BR[16][4][32];   // Wr pairs
    __shared__ v2f BI[16][4][32];   // Wi pairs
    __shared__ v2f BS[16][4][32];   // (Wr+Wi) pairs

    const int tid  = threadIdx.x;
    const int lane = tid & 31;
    const int wib  = tid >> 5;            // wave in block, 0..7

    const int g         = blockIdx.x >> 7;    // 128 blocks per group
    const int row_block = blockIdx.x & 127;   // 8 row-tiles (of 16 rows) each

    const int ll = lane & 15;             // M row (A) / N col (B,C,D)
    const int lh = lane >> 4;             // K-pair half select

    // ---- Stage W_g into LDS, deinterleaved + Gauss-sum precomputed ----
    // B-fragment layout (f32, K=4): lane ll holds column o; VGPR0/1 = K={0,1}
    // for lanes 0-15, K={2,3} for lanes 16-31.
    const float* wg = w + (size_t)g * CG_BLK * CG_BLK * 2;
#pragma unroll
    for (int e = tid; e < 2048; e += 256) {
        const int l  = e & 31;
        const int nt = (e >> 5) & 3;
        const int ks = e >> 7;
        const int i0 = ks * 4 + 2 * (l >> 4);
        const int o  = nt * 16 + (l & 15);
        v2f w0 = *(const v2f*)(wg + ((size_t)i0       * CG_BLK + o) * 2); // {wr,wi}(i0,  o)
        v2f w1 = *(const v2f*)(wg + ((size_t)(i0 + 1) * CG_BLK + o) * 2); // {wr,wi}(i0+1,o)
        BR[ks][nt][l] = (v2f){w0.x, w1.x};
        BI[ks][nt][l] = (v2f){w0.y, w1.y};
        BS[ks][nt][l] = (v2f){w0.x + w0.y, w1.x + w1.y};
    }
    __syncthreads();

    // ---- Main GEMM: this wave owns 16 rows x all 64 outputs of group g ----
    const int row_tile = row_block * 8 + wib;          // 0..1023
    const size_t arow  = (size_t)(row_tile * 16 + ll) * CG_RSTRIDE
                       + (size_t)g * CG_BLK + 2 * lh;
    const float* xr_base = xr + arow;
    const float* xi_base = xi + arow;

    v8f p1[4] = {v8f{}, v8f{}, v8f{}, v8f{}};
    v8f p2[4] = {v8f{}, v8f{}, v8f{}, v8f{}};
    v8f p3[4] = {v8f{}, v8f{}, v8f{}, v8f{}};

#pragma unroll
    for (int ks = 0; ks < 16; ++ks) {
        const int k4 = ks * 4;
        v2f ar = *(const v2f*)(xr_base + k4);
        v2f ai = *(const v2f*)(xi_base + k4);
        v2f as = ar + ai;

#pragma unroll
        for (int nt = 0; nt < 4; ++nt) {
            v2f br = BR[ks][nt][lane];
            v2f bi = BI[ks][nt][lane];
            v2f bs = BS[ks][nt][lane];
            p1[nt] = __builtin_amdgcn_wmma_f32_16x16x4_f32(
                false, ar, false, br, (short)0, p1[nt], false, false);
            p2[nt] = __builtin_amdgcn_wmma_f32_16x16x4_f32(
                false, ai, false, bi, (short)0, p2[nt], false, false);
            p3[nt] = __builtin_amdgcn_wmma_f32_16x16x4_f32(
                false, as, false, bs, (short)0, p3[nt], false, false);
        }
    }

    // ---- Epilogue: Yr = P1-P2, Yi = P3-P1-P2, store interleaved (re,im) b64 ----
    // C/D layout: VGPR v, lanes 0-15 -> M=v, lanes 16-31 -> M=v+8; N = lane%16.
    float* outp = out + ((size_t)(row_tile * 16) * CG_RSTRIDE + (size_t)g * CG_BLK) * 2;
#pragma unroll
    for (int nt = 0; nt < 4; ++nt) {
        const int o = nt * 16 + ll;
#pragma unroll
        for (int v = 0; v < 8; ++v) {
            const int m = v + 8 * lh;
            const float a = p1[nt][v], b = p2[nt][v], c = p3[nt][v];
            v2f val = {a - b, c - a - b};
            *(v2f*)(outp + ((size_t)m * CG_RSTRIDE + o) * 2) = val;
        }
    }
}

extern "C" void kernel_launch(void* const* d_in, const int* in_sizes, int n_in,
                              void* d_out, int out_size, void* d_ws, size_t ws_size,
                              hipStream_t stream) {
    const float* xr = (const float*)d_in[0];   // [4,4096,32,64] f32
    const float* xi = (const float*)d_in[1];   // [4,4096,32,64] f32
    const float* w  = (const float*)d_in[2];   // [32,64,64,2]   f32
    float* out      = (float*)d_out;           // [4,4096,32,64,2] f32

    // 4096 blocks x 256 threads: block -> (group, 8 row-tiles), wave -> 16 rows.
    const int blocks = (CG_ROWS / 16) * CG_GROUPS / 8;   // 4096
    ComplexGroupedMLP_70592082477333_kernel<<<blocks, 256, 0, stream>>>(xr, xi, w, out);
}